// HALTON_33277406609678
// MI455X (gfx1250) — compile-verified
//
#include <hip/hip_runtime.h>

#define B_ 64
#define L_ 512
#define D_ 768
#define H_ 768
#define K_ 128

typedef float v2f __attribute__((ext_vector_type(2)));
typedef float v8f __attribute__((ext_vector_type(8)));

// ---------------------------------------------------------------------------
// Kernel 1: compaction + span-max feature.  One 512-thread block per batch.
// feat[b,d] = max over j in [lo,hi] of (j < nvalid ? enc[b, idx_j, d] : 0)
// where idx_j is the row index of the j-th valid entry (valid_mask == 1).
// ---------------------------------------------------------------------------
__global__ __launch_bounds__(512) void feat_kernel(
    const float* __restrict__ enc,
    const int*   __restrict__ valid_mask,
    const int*   __restrict__ pos_span,
    float*       __restrict__ feat)
{
    __shared__ int s_scan[L_];
    __shared__ int s_idx[64];
    __shared__ int s_n;

    const int b   = blockIdx.x;
    const int tid = threadIdx.x;
    const int* vm = valid_mask + (size_t)b * L_;

    const int v = (vm[tid] == 1) ? 1 : 0;
    s_scan[tid] = v;
    __syncthreads();

    // Hillis-Steele inclusive scan over 512 elements.
    for (int off = 1; off < L_; off <<= 1) {
        int add = (tid >= off) ? s_scan[tid - off] : 0;
        __syncthreads();
        s_scan[tid] += add;
        __syncthreads();
    }

    const int pos = s_scan[tid] - 1;          // compacted position of this row
    if (v && pos < 64) s_idx[pos] = tid;      // only first 64 ever needed (hi<40)
    if (tid == 0) {
        int nv = s_scan[L_ - 1];
        s_n = (nv < 64) ? nv : 64;
    }
    __syncthreads();

    const int lo = pos_span[b * 2 + 0];
    const int hi = pos_span[b * 2 + 1];
    const int nv = s_n;

    for (int d = tid; d < D_; d += 512) {
        float acc = -__builtin_inff();
        for (int j = lo; j <= hi; ++j) {
            float val = (j < nv)
                ? enc[((size_t)b * L_ + s_idx[j]) * D_ + d]
                : 0.0f;                        // zero rows past the valid count
            acc = fmaxf(acc, val);
        }
        feat[(size_t)b * D_ + d] = acc;
    }
}

// ---------------------------------------------------------------------------
// Kernel 2/3: C(64,N) = act(A(64,Kd) @ Bm(Kd,N) + bias)  via
// V_WMMA_F32_16X16X4_F32 (exact f32, matches the f32 reference).
// Block = 128 threads (4 waves): wave w owns M-tile [16w,16w+16),
// blockIdx.x owns N-tile [16*bx, 16*bx+16).  K marched in 64-wide panels
// staged through LDS (shared by all 4 waves, bank-conflict free).
// ---------------------------------------------------------------------------
template <bool RELU>
__global__ __launch_bounds__(128) void gemm_wmma(
    const float* __restrict__ A,
    const float* __restrict__ Bm,
    const float* __restrict__ bias,
    float*       __restrict__ C,
    int N, int Kd)
{
    __shared__ float sB[64 * 16];             // 4 KB K-panel of B

    const int lane = threadIdx.x & 31;
    const int wave = threadIdx.x >> 5;
    const int n0   = blockIdx.x * 16;
    const int m0   = wave * 16;
    const int m    = lane & 15;               // A row within tile
    const int n    = lane & 15;               // B/C column within tile
    const int kb   = (lane >> 4) * 2;         // half-wave K offset (0 or 2)

    v8f acc = {};
    const float* arow   = A + (size_t)(m0 + m) * Kd;
    const float  bias_n = bias[n0 + n];

    for (int kc = 0; kc < Kd; kc += 64) {
        // Stage B[kc:kc+64, n0:n0+16] into LDS (128 threads, 8 elems each).
        for (int i = threadIdx.x; i < 64 * 16; i += 128) {
            sB[i] = Bm[(size_t)(kc + (i >> 4)) * N + n0 + (i & 15)];
        }
        __syncthreads();

        #pragma unroll
        for (int k = 0; k < 64; k += 4) {
            v2f a;                            // A 16x4 frag: 2 consecutive K
            a.x = arow[kc + k + kb];
            a.y = arow[kc + k + kb + 1];
            v2f bf;                           // B 4x16 frag from LDS
            bf.x = sB[(k + kb)     * 16 + n];
            bf.y = sB[(k + kb + 1) * 16 + n];
            // (neg_a, A, neg_b, B, c_mod, C, reuse_a, reuse_b)
            acc = __builtin_amdgcn_wmma_f32_16x16x4_f32(
                false, a, false, bf, (short)0, acc, false, false);
        }
        __syncthreads();
    }

    // C/D layout: VGPR v holds M = 8*(lane>>4) + v, N = lane&15.
    const int mrow = m0 + (lane >> 4) * 8;
    #pragma unroll
    for (int vv = 0; vv < 8; ++vv) {
        float val = acc[vv] + bias_n;
        if (RELU) val = fmaxf(val, 0.0f);
        C[(size_t)(mrow + vv) * N + n0 + n] = val;
    }
}

// ---------------------------------------------------------------------------
extern "C" void kernel_launch(void* const* d_in, const int* in_sizes, int n_in,
                              void* d_out, int out_size, void* d_ws, size_t ws_size,
                              hipStream_t stream)
{
    (void)in_sizes; (void)n_in; (void)out_size; (void)ws_size;

    const float* enc        = (const float*)d_in[0];   // (B,L,D) f32
    const float* W1         = (const float*)d_in[1];   // (D,H)   f32
    const float* b1         = (const float*)d_in[2];   // (H,)    f32
    const float* W2         = (const float*)d_in[3];   // (H,K)   f32
    const float* b2         = (const float*)d_in[4];   // (K,)    f32
    const int*   valid_mask = (const int*)d_in[5];     // (B,L)   i32
    const int*   pos_span   = (const int*)d_in[6];     // (B,2)   i32
    // d_in[7] = mask_span: unused (mask_feat is discarded by reference)

    float* out  = (float*)d_out;                       // (B,K) f32
    float* feat = (float*)d_ws;                        // (B,D) = 192 KB
    float* h    = feat + (size_t)B_ * D_;              // (B,H) = 192 KB

    feat_kernel<<<B_, 512, 0, stream>>>(enc, valid_mask, pos_span, feat);
    gemm_wmma<true ><<<H_ / 16, 128, 0, stream>>>(feat, W1, b1, h,   H_, D_);
    gemm_wmma<false><<<K_ / 16, 128, 0, stream>>>(h,    W2, b2, out, K_, H_);
}